// GINE_28398323761988
// MI455X (gfx1250) — compile-verified
//
#include <hip/hip_runtime.h>
#include <hip/hip_bf16.h>

// ---------------------------------------------------------------------------
// GINE forward on gfx1250 (MI455X), full fp32 via V_WMMA_F32_16X16X4_F32.
// Edge pipeline fused (edge_lin GEMM + gather + relu + atomic scatter) so the
// [E,cin] message tensor never touches HBM; agg[N,cin] (<=64MB) lives in L2.
// Weights are register-resident in the edge kernel (10 tiles/wave) and shared
// across 5 node tiles in the node kernel to maximize WMMA density per load.
// ---------------------------------------------------------------------------

typedef __attribute__((ext_vector_type(2))) float v2f;
typedef __attribute__((ext_vector_type(8))) float v8f;

__device__ __forceinline__ v8f wmma_f32(v2f a, v2f b, v8f c) {
    // 8 args: (neg_a, A, neg_b, B, c_mod, C, reuse_a, reuse_b)
    return __builtin_amdgcn_wmma_f32_16x16x4_f32(false, a, false, b, (short)0, c,
                                                 false, false);
}

#define N_NODES 50000
#define N_EDGES 800000
#define N_GRAPHS 500
#define GINE_EPS 1e-5f
#define RSQRT2 0.70710678118654752f
#define ETILES 10   // edge tiles per wave in k_msg_agg (50000 % 10 == 0)
#define MTILES 5    // node tiles per wave in k_node_update (3125 % 5 == 0)

// ---------------------------------------------------------------------------
// vert_emb: x[N,13] @ vert_w[13,64] -> x0[N,64].  K=13 (not /4): tiny, VALU.
// ---------------------------------------------------------------------------
__global__ void k_vert_emb(const float* __restrict__ x,
                           const float* __restrict__ w,
                           float* __restrict__ out) {
    int idx = blockIdx.x * blockDim.x + threadIdx.x;
    if (idx >= N_NODES * 64) return;
    int node = idx >> 6, c = idx & 63;
    float s = 0.f;
#pragma unroll
    for (int k = 0; k < 13; ++k) s = fmaf(x[node * 13 + k], w[k * 64 + c], s);
    out[node * 64 + c] = s;
}

// ---------------------------------------------------------------------------
// edge_emb: edge_attr[E,4] @ edge_w[4,64] -> e[E,64].
// Exactly the native 16x16x4 f32 WMMA shape: one WMMA per 16x16 tile.
// Block = 128 threads = 4 waves, each wave covers one 16-col chunk of 64.
// ---------------------------------------------------------------------------
__global__ void k_edge_emb(const float* __restrict__ ea,
                           const float* __restrict__ ew,
                           float* __restrict__ e) {
    int tile = blockIdx.x;
    int wave = threadIdx.x >> 5;          // n-chunk 0..3
    int lane = threadIdx.x & 31;
    int l16 = lane & 15, kh = lane >> 4;  // kh selects K pair {0,1} or {2,3}
    int row = tile * 16 + l16;            // A: M = lane%16
    int n = wave * 16 + l16;              // B/D: N = lane%16
    v2f a, b;
    v8f acc = {};
    a.x = ea[row * 4 + 2 * kh + 0];
    a.y = ea[row * 4 + 2 * kh + 1];
    b.x = ew[(2 * kh + 0) * 64 + n];
    b.y = ew[(2 * kh + 1) * 64 + n];
    acc = wmma_f32(a, b, acc);
#pragma unroll
    for (int r = 0; r < 8; ++r)           // D: M = r + 8*(lane/16)
        e[(tile * 16 + 8 * kh + r) * 64 + n] = acc[r];
}

// ---------------------------------------------------------------------------
// Fused edge message + aggregate for one layer:
//   agg[dst] += relu( x[src] + e @ edge_lin_w )
// One wave per block. blockIdx.y = 16-wide cin chunk; the wave preloads ALL
// 16 B fragments for its chunk into registers (32 VGPRs), then sweeps ETILES
// 16-edge tiles: per tile 16 b64 A-loads + 16 WMMAs + gather/relu/atomic
// epilogue. agg fits in L2 so the atomics resolve on-chip.
// ---------------------------------------------------------------------------
__global__ void k_msg_agg(const float* __restrict__ e,    // [E,64]
                          const float* __restrict__ w,    // [64,cin]
                          const float* __restrict__ x,    // [N,cin]
                          const int* __restrict__ src,
                          const int* __restrict__ dst,
                          float* __restrict__ agg,        // [N,cin] (zeroed)
                          int cin) {
    int lane = threadIdx.x & 31;
    int l16 = lane & 15, kh = lane >> 4;
    int n = blockIdx.y * 16 + l16;

    // Register-resident B fragments for this chunk (K = 0..63).
    v2f bf[16];
#pragma unroll
    for (int kk = 0; kk < 16; ++kk) {
        int k0 = kk * 4 + 2 * kh;
        bf[kk].x = w[(k0 + 0) * cin + n];
        bf[kk].y = w[(k0 + 1) * cin + n];
    }

    int tile0 = blockIdx.x * ETILES;
    const float* erow = e + (size_t)(tile0 * 16 + l16) * 64 + 2 * kh;
    for (int t = 0; t < ETILES; ++t) {
        int tile = tile0 + t;
        __builtin_prefetch(erow + 16 * 64, 0, 1);  // next tile's row
        v8f acc = {};
#pragma unroll
        for (int kk = 0; kk < 16; ++kk) {
            v2f a;
            a.x = erow[kk * 4 + 0];
            a.y = erow[kk * 4 + 1];
            acc = wmma_f32(a, bf[kk], acc);
        }
#pragma unroll
        for (int r = 0; r < 8; ++r) {
            int eid = tile * 16 + 8 * kh + r;
            int s = src[eid], d = dst[eid];
            float v = acc[r] + x[(size_t)s * cin + n];
            v = fmaxf(v, 0.f);
            atomicAdd(&agg[(size_t)d * cin + n], v);
        }
        erow += 16 * 64;
    }
}

// ---------------------------------------------------------------------------
// Fused node update:
//   h  = relu( ((1+eps)*x + agg) @ lin_w + lin_b )
//   xo = ( h + x @ res_w ) / sqrt(2)
// One wave handles MTILES consecutive 16-row tiles for one 16-col chunk so
// each lin_w/res_w fragment is reused by MTILES WMMAs.
// ---------------------------------------------------------------------------
__global__ void k_node_update(const float* __restrict__ x,    // [N,cin]
                              const float* __restrict__ agg,  // [N,cin]
                              const float* __restrict__ lw,   // [cin,cout]
                              const float* __restrict__ lb,   // [cout]
                              const float* __restrict__ rw,   // [cin,cout]
                              float* __restrict__ xo,         // [N,cout]
                              int cin, int cout) {
    int mt0 = blockIdx.x * MTILES, nt = blockIdx.y;
    int lane = threadIdx.x & 31;
    int l16 = lane & 15, kh = lane >> 4;
    int n = nt * 16 + l16;

    const float* xr[MTILES];
    const float* gr[MTILES];
#pragma unroll
    for (int m = 0; m < MTILES; ++m) {
        size_t row = (size_t)(mt0 + m) * 16 + l16;
        xr[m] = x + row * cin + 2 * kh;
        gr[m] = agg + row * cin + 2 * kh;
    }

    v8f acc1[MTILES], acc2[MTILES];
#pragma unroll
    for (int m = 0; m < MTILES; ++m) { acc1[m] = (v8f){}; acc2[m] = (v8f){}; }

    for (int k0 = 0; k0 < cin; k0 += 4) {
        v2f b1, b2;
        b1.x = lw[(k0 + 2 * kh + 0) * cout + n];
        b1.y = lw[(k0 + 2 * kh + 1) * cout + n];
        b2.x = rw[(k0 + 2 * kh + 0) * cout + n];
        b2.y = rw[(k0 + 2 * kh + 1) * cout + n];
#pragma unroll
        for (int m = 0; m < MTILES; ++m) {
            float x0 = xr[m][k0], x1 = xr[m][k0 + 1];
            v2f a1, a2;
            a2.x = x0;
            a2.y = x1;
            a1.x = (1.f + GINE_EPS) * x0 + gr[m][k0];
            a1.y = (1.f + GINE_EPS) * x1 + gr[m][k0 + 1];
            acc1[m] = wmma_f32(a1, b1, acc1[m]);
            acc2[m] = wmma_f32(a2, b2, acc2[m]);
        }
    }
    float bias = lb[n];
#pragma unroll
    for (int m = 0; m < MTILES; ++m) {
#pragma unroll
        for (int r = 0; r < 8; ++r) {
            int row = (mt0 + m) * 16 + 8 * kh + r;
            float h = fmaxf(acc1[m][r] + bias, 0.f);
            xo[(size_t)row * cout + n] = (h + acc2[m][r]) * RSQRT2;
        }
    }
}

// ---------------------------------------------------------------------------
// Global mean pool: sums/counts via L2-resident atomics, then divide.
// ---------------------------------------------------------------------------
__global__ void k_pool_sum(const float* __restrict__ x,
                           const int* __restrict__ batch,
                           float* __restrict__ sums,
                           float* __restrict__ cnt, int width) {
    long long idx = (long long)blockIdx.x * blockDim.x + threadIdx.x;
    if (idx >= (long long)N_NODES * width) return;
    int node = (int)(idx / width), c = (int)(idx % width);
    int g = batch[node];
    atomicAdd(&sums[(size_t)g * width + c], x[(size_t)node * width + c]);
    if (c == 0) atomicAdd(&cnt[g], 1.f);
}

__global__ void k_pool_div(const float* __restrict__ sums,
                           const float* __restrict__ cnt,
                           float* __restrict__ pooled, int width) {
    int idx = blockIdx.x * blockDim.x + threadIdx.x;
    if (idx >= N_GRAPHS * width) return;
    pooled[idx] = sums[idx] / fmaxf(cnt[idx / width], 1.f);
}

// ---------------------------------------------------------------------------
// Head GEMM 1: pooled[500,448] @ w1[448,512] + b1 -> gelu(tanh) -> hid[500,512]
// Row bounds handled with selects (EXEC stays all-ones for WMMA).
// ---------------------------------------------------------------------------
__global__ void k_head1(const float* __restrict__ p,
                        const float* __restrict__ w1,
                        const float* __restrict__ b1,
                        float* __restrict__ hid) {
    int mt = blockIdx.x, nt = blockIdx.y;
    int lane = threadIdx.x & 31;
    int l16 = lane & 15, kh = lane >> 4;
    int row = mt * 16 + l16;
    bool ok = row < N_GRAPHS;
    int n = nt * 16 + l16;
    v8f acc = {};
    for (int k0 = 0; k0 < 448; k0 += 4) {
        v2f a, b;
        a.x = ok ? p[(size_t)row * 448 + k0 + 2 * kh + 0] : 0.f;
        a.y = ok ? p[(size_t)row * 448 + k0 + 2 * kh + 1] : 0.f;
        b.x = w1[(k0 + 2 * kh + 0) * 512 + n];
        b.y = w1[(k0 + 2 * kh + 1) * 512 + n];
        acc = wmma_f32(a, b, acc);
    }
    float bias = b1[n];
#pragma unroll
    for (int r = 0; r < 8; ++r) {
        int m = mt * 16 + 8 * kh + r;
        if (m < N_GRAPHS) {
            float v = acc[r] + bias;
            // jax.nn.gelu default = tanh approximation
            float t = 0.7978845608028654f * (v + 0.044715f * v * v * v);
            hid[(size_t)m * 512 + n] = 0.5f * v * (1.f + tanhf(t));
        }
    }
}

// Head GEMM 2: hid[500,512] @ w2[512] + b2 -> out[500]; one wave per graph.
__global__ void k_head2(const float* __restrict__ hid,
                        const float* __restrict__ w2,
                        const float* __restrict__ b2,
                        float* __restrict__ out) {
    int g = blockIdx.x;
    int lane = threadIdx.x;
    float s = 0.f;
    for (int k = lane; k < 512; k += 32)
        s = fmaf(hid[(size_t)g * 512 + k], w2[k], s);
#pragma unroll
    for (int off = 16; off; off >>= 1) s += __shfl_xor(s, off, 32);
    if (lane == 0) out[g] = s + b2[0];
}

// ---------------------------------------------------------------------------
extern "C" void kernel_launch(void* const* d_in, const int* in_sizes, int n_in,
                              void* d_out, int out_size, void* d_ws, size_t ws_size,
                              hipStream_t stream) {
    (void)in_sizes; (void)n_in; (void)out_size; (void)ws_size;

    const float* x_in   = (const float*)d_in[0];
    const int*   eidx   = (const int*)d_in[1];
    const float* eattr  = (const float*)d_in[2];
    const int*   batch  = (const int*)d_in[3];
    const float* vert_w = (const float*)d_in[4];
    const float* edge_w = (const float*)d_in[5];
    const float* Wel[5]; const float* Wl[5]; const float* Bl[5]; const float* Wr[5];
    for (int i = 0; i < 5; ++i) {
        Wel[i] = (const float*)d_in[6 + 4 * i];
        Wl[i]  = (const float*)d_in[7 + 4 * i];
        Bl[i]  = (const float*)d_in[8 + 4 * i];
        Wr[i]  = (const float*)d_in[9 + 4 * i];
    }
    const float* w1 = (const float*)d_in[26];
    const float* b1 = (const float*)d_in[27];
    const float* w2 = (const float*)d_in[28];
    const float* b2 = (const float*)d_in[29];

    static const int widths[6] = {1, 1, 2, 3, 5, 7};  // *64

    // Workspace carve-up (deterministic offsets each call).
    char* ws = (char*)d_ws;
    size_t off = 0;
    auto carve = [&](size_t bytes) -> void* {
        void* p = ws + off;
        off = (off + bytes + 255) & ~(size_t)255;
        return p;
    };
    float* xA     = (float*)carve((size_t)N_NODES * 320 * 4);  // widths 64,128,320
    float* xB     = (float*)carve((size_t)N_NODES * 448 * 4);  // widths 64,192,448
    float* e      = (float*)carve((size_t)N_EDGES * 64 * 4);
    float* agg    = (float*)carve((size_t)N_NODES * 320 * 4);
    float* sums   = (float*)carve((size_t)N_GRAPHS * 448 * 4);
    float* cnt    = (float*)carve((size_t)N_GRAPHS * 4);
    float* pooled = (float*)carve((size_t)N_GRAPHS * 448 * 4);
    float* hid    = (float*)carve((size_t)N_GRAPHS * 512 * 4);

    const int* src = eidx;            // edge_index[0]
    const int* dst = eidx + N_EDGES;  // edge_index[1]

    // Embeddings
    k_vert_emb<<<(N_NODES * 64 + 255) / 256, 256, 0, stream>>>(x_in, vert_w, xA);
    k_edge_emb<<<N_EDGES / 16, 128, 0, stream>>>(eattr, edge_w, e);

    // 5 GINE layers (E/16 = 50000 tiles, N/16 = 3125 tiles)
    float* xc = xA; float* xn = xB;
    for (int i = 0; i < 5; ++i) {
        int cin = widths[i] * 64, cout = widths[i + 1] * 64;
        hipMemsetAsync(agg, 0, (size_t)N_NODES * cin * 4, stream);
        dim3 eg((N_EDGES / 16) / ETILES, cin / 16);
        k_msg_agg<<<eg, 32, 0, stream>>>(e, Wel[i], xc, src, dst, agg, cin);
        dim3 gg((N_NODES / 16) / MTILES, cout / 16);
        k_node_update<<<gg, 32, 0, stream>>>(xc, agg, Wl[i], Bl[i], Wr[i], xn, cin, cout);
        float* t = xc; xc = xn; xn = t;
    }
    // xc now holds x5 [N, 448]

    // Mean pool + head
    hipMemsetAsync(sums, 0, (size_t)N_GRAPHS * 448 * 4, stream);
    hipMemsetAsync(cnt, 0, (size_t)N_GRAPHS * 4, stream);
    long long pool_threads = (long long)N_NODES * 448;
    k_pool_sum<<<(unsigned)((pool_threads + 255) / 256), 256, 0, stream>>>(xc, batch, sums, cnt, 448);
    k_pool_div<<<(N_GRAPHS * 448 + 255) / 256, 256, 0, stream>>>(sums, cnt, pooled, 448);
    dim3 hg((N_GRAPHS + 15) / 16, 512 / 16);
    k_head1<<<hg, 32, 0, stream>>>(pooled, w1, b1, hid);
    k_head2<<<N_GRAPHS, 32, 0, stream>>>(hid, w2, b2, (float*)d_out);
}